// FIRFilter_47614007443841
// MI455X (gfx1250) — compile-verified
//
#include <hip/hip_runtime.h>

typedef float v2f __attribute__((ext_vector_type(2)));
typedef float v8f __attribute__((ext_vector_type(8)));

namespace {
constexpr int kNSamples      = 4000000;
constexpr int kNFrames       = 250;
constexpr int kFrame         = 16000;          // samples per frame
constexpr int kNBins         = 1025;
constexpr int kTaps          = 257;            // causal FIR length
constexpr int kHStride       = 260;            // padded tap stride in workspace
constexpr int kTile          = 256;            // outputs per wave tile (16x16)
constexpr int kTilesPerBatch = kNSamples / kTile;  // 15625 (exact)
constexpr int kKDim          = 272;            // 17*16, zero-padded K for GEMM
constexpr int kBatch         = 4;
// LDS layout per wave (floats): A0 | A1 | Hr0 | Hr1
constexpr int kASize = 512 + 2 * 32;           // pad +2 per 16 -> 576, b64-aligned reads
constexpr int kHSize = 288;                    // reversed taps w/ zero guard bands
constexpr int kWaveLds = 2 * kASize + 2 * kHSize;  // 1728 floats = 6912 B
}

// ---------------------------------------------------------------------------
// Kernel 1: h[f][k] = irfft(clip(mags[f]))[(k-127) mod 2048] * hann257[k+2]
// irfft via exact 1025-term cosine sum using a 2048-entry LDS cosine table.
// ---------------------------------------------------------------------------
__global__ __launch_bounds__(256) void build_ir_kernel(const float* __restrict__ mags,
                                                       float* __restrict__ h)
{
    __shared__ float tab[2048];
    __shared__ float mg[kNBins];
    const int f   = blockIdx.x;
    const int tid = threadIdx.x;

    for (int r = tid; r < 2048; r += 256)
        tab[r] = cosf((float)r * 3.0679615757712823e-3f);   // pi/1024
    for (int j = tid; j < kNBins; j += 256) {
        float m = mags[f * kNBins + j];
        mg[j] = fminf(fmaxf(m, 0.0f), 1.0f);                // clip_by_value
    }
    __syncthreads();

    for (int k = tid; k < kTaps; k += 256) {
        float val = 0.0f;
        if (k < 255) {                                      // taps 255,256 windowed to 0
            const int t = (k < 127) ? (k + 1921) : (k - 127);
            float s = 0.0f;
            for (int j = 1; j < 1024; ++j)
                s = fmaf(mg[j], tab[(j * t) & 2047], s);
            const float ir = mg[0] + ((t & 1) ? -mg[1024] : mg[1024]) + 2.0f * s;
            // periodic Hann window of size 257, index k+2
            const float w = 0.5f - 0.5f * cosf(6.283185307179586f * (float)(k + 2) *
                                               (1.0f / 257.0f));
            val = ir * (1.0f / 2048.0f) * w;
        }
        h[f * kHStride + k] = val;
    }
}

// ---------------------------------------------------------------------------
// Kernel 2: time-varying FIR as Toeplitz GEMM on V_WMMA_F32_16X16X4_F32.
// One wave per 256-output tile; fully unrolled branch-free K-loop of 68 WMMAs
// with immediate-offset DS loads; boundary tiles run a second pre-masked pass.
// ---------------------------------------------------------------------------
__global__ __launch_bounds__(256) void fir_wmma_kernel(const float* __restrict__ audio,
                                                       const float* __restrict__ h,
                                                       float* __restrict__ out)
{
    __shared__ float lds[8][kWaveLds];
    const int lane   = threadIdx.x & 31;
    const int wave   = threadIdx.x >> 5;
    const int tileId = blockIdx.x * 8 + wave;
    if (tileId >= kBatch * kTilesPerBatch) return;          // wave-uniform exit

    const int b  = tileId / kTilesPerBatch;
    const int n0 = (tileId - b * kTilesPerBatch) * kTile;

    float* __restrict__ A0  = &lds[wave][0];
    float* __restrict__ A1  = A0 + kASize;
    float* __restrict__ Hr0 = A1 + kASize;
    float* __restrict__ Hr1 = Hr0 + kHSize;

    const float* __restrict__ aud = audio + (long)b * kNSamples;
    const int sBase = n0 - 129;                             // first audio sample of A

    const int fLo   = max(sBase, 0) / kFrame;
    const int fHi   = min(n0 + 382, kNSamples - 1) / kFrame;   // fLo or fLo+1
    const int nPass = (fHi > fLo) ? 2 : 1;
    const int cut   = fHi * kFrame;                         // pass split sample

    // Stage pass-0 audio window (bounds-zeroed; boundary tiles also mask s>=cut).
    for (int idx = lane; idx < 512; idx += 32) {
        const int s = sBase + idx;
        float v = (s >= 0 && s < kNSamples) ? aud[s] : 0.0f;
        if (nPass == 2 && s >= cut) v = 0.0f;
        A0[idx + ((idx >> 4) << 1)] = v;
    }
    // Stage REVERSED guard-banded taps: Hr[q] = h[271-q] (zero outside [15,271]).
    for (int q = lane; q < kHSize; q += 32) {
        const int t = 271 - q;
        Hr0[q] = ((unsigned)t <= 256u) ? h[fLo * kHStride + t] : 0.0f;
    }
    if (nPass == 2) {                                       // rare: ~4/125 of tiles
        for (int idx = lane; idx < 512; idx += 32) {
            const int s = sBase + idx;
            const float v = (s >= cut && s < kNSamples) ? aud[s] : 0.0f;
            A1[idx + ((idx >> 4) << 1)] = v;
        }
        for (int q = lane; q < kHSize; q += 32) {
            const int t = 271 - q;
            Hr1[q] = ((unsigned)t <= 256u) ? h[fHi * kHStride + t] : 0.0f;
        }
    }
    // Wave-private LDS: draining the DS counter is sufficient (no block barrier).
    asm volatile("s_wait_dscnt 0x0" ::: "memory");

    v8f acc = {0.f, 0.f, 0.f, 0.f, 0.f, 0.f, 0.f, 0.f};
    const int M    = lane & 15;                             // A row / B column
    const int kOff = (lane >> 4) << 1;                      // K half-split: 0 or 2

    for (int pass = 0; pass < nPass; ++pass) {
        // Per-lane base pointers; all per-step offsets become immediates below.
        const float* __restrict__ aB = (pass ? A1 : A0) + 18 * M + kOff;
        const float* __restrict__ bB = (pass ? Hr1 : Hr0) + (15 - M) + kOff;
#pragma unroll
        for (int kk0 = 0; kk0 < kKDim; kk0 += 4) {
            const int aOff = kk0 + ((kk0 >> 4) << 1);       // compile-time constant
            const v2f av = *(const v2f*)(aB + aOff);        // ds_load_b64, imm offset
            v2f bv;
            bv.x = bB[kk0];                                 // ascending pair: no swaps
            bv.y = bB[kk0 + 1];
            acc = __builtin_amdgcn_wmma_f32_16x16x4_f32(false, av, false, bv,
                                                        (short)0, acc, false, false);
        }
    }

    // C layout: VGPR r -> lanes 0-15: (row r, col=lane); lanes 16-31: (row r+8).
    float* __restrict__ o = out + (long)b * kNSamples + n0;
    const int rowOff = (lane >> 4) << 3;
#pragma unroll
    for (int r = 0; r < 8; ++r)
        o[((r + rowOff) << 4) + M] = acc[r];
}

// ---------------------------------------------------------------------------
extern "C" void kernel_launch(void* const* d_in, const int* in_sizes, int n_in,
                              void* d_out, int out_size, void* d_ws, size_t ws_size,
                              hipStream_t stream)
{
    (void)in_sizes; (void)n_in; (void)out_size; (void)ws_size;
    const float* audio = (const float*)d_in[0];   // [4, 4000000] f32
    const float* mags  = (const float*)d_in[1];   // [1, 250, 1025] f32
    float* out = (float*)d_out;                   // [4, 4000000] f32
    float* h   = (float*)d_ws;                    // 250 * 260 * 4 B = 260 KB scratch

    build_ir_kernel<<<kNFrames, 256, 0, stream>>>(mags, h);

    const int tiles = kBatch * kTilesPerBatch;    // 62500 wave-tiles
    fir_wmma_kernel<<<(tiles + 7) / 8, 256, 0, stream>>>(audio, h, out);
}